// GCNEncoder_5995774345966
// MI455X (gfx1250) — compile-verified
//
#include <hip/hip_runtime.h>

typedef float v2f __attribute__((ext_vector_type(2)));
typedef float v8f __attribute__((ext_vector_type(8)));

#define DIMK 128   // feature dim (D == H == 128)

// ---------------------------------------------------------------------------
// 1) zero scratch (vectorized)
// ---------------------------------------------------------------------------
__global__ void sage_zero4(float4* __restrict__ p, long n4) {
    long i = (long)blockIdx.x * blockDim.x + threadIdx.x;
    if (i < n4) p[i] = make_float4(0.f, 0.f, 0.f, 0.f);
}

// ---------------------------------------------------------------------------
// 2) edge gather + scatter-add: one wave32 per edge, float4 per lane
// ---------------------------------------------------------------------------
__global__ void sage_scatter(const float* __restrict__ x,
                             const int* __restrict__ src,
                             const int* __restrict__ dst,
                             float* __restrict__ agg,
                             float* __restrict__ cnt,
                             int E, int addCnt) {
    long gid = (long)blockIdx.x * blockDim.x + threadIdx.x;
    int e    = (int)(gid >> 5);
    int lane = threadIdx.x & 31;
    if (e >= E) return;
    int s = src[e];
    int d = dst[e];
    const float4* xs = (const float4*)(x + (long)s * DIMK);
    float4 v = xs[lane];                       // 32 lanes * 4 = 128 floats
    float* ag = agg + (long)d * DIMK + lane * 4;
    atomicAdd(ag + 0, v.x);
    atomicAdd(ag + 1, v.y);
    atomicAdd(ag + 2, v.z);
    atomicAdd(ag + 3, v.w);
    if (addCnt && lane == 0) atomicAdd(cnt + d, 1.0f);
}

// ---------------------------------------------------------------------------
// 3) inv[i] = 1 / max(cnt[i], 1)
// ---------------------------------------------------------------------------
__global__ void sage_invcnt(const float* __restrict__ cnt,
                            float* __restrict__ inv, int n) {
    int i = blockIdx.x * blockDim.x + threadIdx.x;
    if (i < n) inv[i] = 1.0f / fmaxf(cnt[i], 1.0f);
}

// ---------------------------------------------------------------------------
// 4) fused SAGE GEMM with fp32 WMMA:
//    out = (agg .* inv) @ Wl^T + xin @ Wr^T + bias   (+ optional PReLU)
//    block = 256 threads = 8 waves; each wave owns a 16x128 output strip.
// ---------------------------------------------------------------------------
__global__ __launch_bounds__(256) void sage_gemm_wmma(
        const float* __restrict__ agg, const float* __restrict__ inv,
        const float* __restrict__ xin,
        const float* __restrict__ Wl,  const float* __restrict__ Wr,
        const float* __restrict__ bias, const float* __restrict__ prelu,
        float* __restrict__ out, int N, int applyPrelu) {
    // Stage W^T in LDS with pitch 129 (bank-conflict-free stores & reads).
    __shared__ float sWl[DIMK * 129];
    __shared__ float sWr[DIMK * 129];
    for (int idx = threadIdx.x; idx < DIMK * DIMK; idx += 256) {
        int n = idx >> 7;        // row of W
        int k = idx & 127;       // col of W
        sWl[k * 129 + n] = Wl[idx];   // sW[k][n] = W[n][k]
        sWr[k * 129 + n] = Wr[idx];
    }
    __syncthreads();

    int wave = threadIdx.x >> 5;
    int lane = threadIdx.x & 31;
    int row0 = (blockIdx.x * 8 + wave) * 16;
    if (row0 >= N) return;                 // whole wave exits: EXEC stays all-1 for WMMA

    int half = lane >> 4;                  // 0: lanes 0-15, 1: lanes 16-31
    int l16  = lane & 15;
    int r = row0 + l16;
    if (r >= N) r = N - 1;                 // clamp (N%16==0 here, but stay safe)
    float invr = inv[r];
    const float* aggr = agg + (long)r * DIMK;
    const float* xr   = xin + (long)r * DIMK;

    v8f acc[8];
#pragma unroll
    for (int t = 0; t < 8; ++t) acc[t] = (v8f)(0.0f);

    // K loop: 16x16x4 fp32 WMMA, A = activations (16x4), B = W^T slab (4x16)
    for (int k = 0; k < DIMK; k += 4) {
        // A fragment: lane half 0 holds K=k,k+1 ; half 1 holds K=k+2,k+3
        int ka = k + half * 2;
        v2f aM, aX;
        aM.x = aggr[ka] * invr;  aM.y = aggr[ka + 1] * invr;
        aX.x = xr[ka];           aX.y = xr[ka + 1];

        int kb0 = (k + half) * 129;        // B VGPR0: rows K=k (lo), k+1 (hi)
        int kb1 = (k + 2 + half) * 129;    // B VGPR1: rows K=k+2 (lo), k+3 (hi)
#pragma unroll
        for (int t = 0; t < 8; ++t) {
            int c = t * 16 + l16;
            v2f bl, br;
            bl.x = sWl[kb0 + c];  bl.y = sWl[kb1 + c];
            br.x = sWr[kb0 + c];  br.y = sWr[kb1 + c];
            acc[t] = __builtin_amdgcn_wmma_f32_16x16x4_f32(
                         false, aM, false, bl, (short)0, acc[t], false, false);
            acc[t] = __builtin_amdgcn_wmma_f32_16x16x4_f32(
                         false, aX, false, br, (short)0, acc[t], false, false);
        }
    }

    float slope = applyPrelu ? *prelu : 0.0f;
#pragma unroll
    for (int t = 0; t < 8; ++t) {
        int c  = t * 16 + l16;
        float bv = bias[c];
#pragma unroll
        for (int v = 0; v < 8; ++v) {
            int row = row0 + v + half * 8;     // C/D layout: VGPR v -> M=v / v+8
            if (row < N) {
                float hv = acc[t][v] + bv;
                if (applyPrelu) hv = (hv >= 0.0f) ? hv : slope * hv;
                out[(long)row * DIMK + c] = hv;
            }
        }
    }
}

// ---------------------------------------------------------------------------
// launcher
// ---------------------------------------------------------------------------
extern "C" void kernel_launch(void* const* d_in, const int* in_sizes, int n_in,
                              void* d_out, int out_size, void* d_ws, size_t ws_size,
                              hipStream_t stream) {
    const float* x   = (const float*)d_in[0];
    const int*   ei  = (const int*)  d_in[1];
    const float* W1l = (const float*)d_in[2];
    const float* b1  = (const float*)d_in[3];
    const float* W1r = (const float*)d_in[4];
    const float* W2l = (const float*)d_in[5];
    const float* b2  = (const float*)d_in[6];
    const float* W2r = (const float*)d_in[7];
    const float* pa  = (const float*)d_in[8];
    float* out = (float*)d_out;

    int N = in_sizes[0] / DIMK;
    int E = in_sizes[1] / 2;
    const int* src = ei;
    const int* dst = ei + E;

    // workspace layout: [agg N*128][cnt N][inv N][h N*128]
    float* agg = (float*)d_ws;
    float* cnt = agg + (size_t)N * DIMK;
    float* inv = cnt + N;
    float* h   = inv + N;

    const int TB = 256;
    long nz1  = (long)N * DIMK + N;               // agg + cnt contiguous
    long nz14 = nz1 / 4;
    long nz2  = (long)N * DIMK;
    long nz24 = nz2 / 4;
    int  zb1  = (int)((nz14 + TB - 1) / TB);
    int  zb2  = (int)((nz24 + TB - 1) / TB);
    int  sb   = (int)(((long)E * 32 + TB - 1) / TB);
    int  ib   = (N + TB - 1) / TB;
    int  gb   = (N + 127) / 128;                  // 8 waves x 16 rows per block

    // ---- layer 1 ----
    sage_zero4  <<<zb1, TB, 0, stream>>>((float4*)agg, nz14);
    sage_scatter<<<sb,  TB, 0, stream>>>(x, src, dst, agg, cnt, E, 1);
    sage_invcnt <<<ib,  TB, 0, stream>>>(cnt, inv, N);
    sage_gemm_wmma<<<gb, TB, 0, stream>>>(agg, inv, x, W1l, W1r, b1, pa, h, N, 1);

    // ---- layer 2 (same degrees; reuse inv) ----
    sage_zero4  <<<zb2, TB, 0, stream>>>((float4*)agg, nz24);
    sage_scatter<<<sb,  TB, 0, stream>>>(h, src, dst, agg, cnt, E, 0);
    sage_gemm_wmma<<<gb, TB, 0, stream>>>(agg, inv, h, W2l, W2r, b2, pa, out, N, 0);
}